// ChamfersDistance3_30511447671630
// MI455X (gfx1250) — compile-verified
//
#include <hip/hip_runtime.h>
#include <math.h>

typedef __attribute__((ext_vector_type(2))) float v2f;
typedef __attribute__((ext_vector_type(8))) float v8f;

// ---------------------------------------------------------------------------
// Pack kernel. For each 3D point z, emit WMMA fragments in *lane order* so the
// hot loop loads them with a single b64 and no per-lane selection.
//
// V_WMMA_F32_16X16X4_F32 fragment layout (wave32): lane l, lo=l&15, hi=l>>4;
// A/B VGPR v holds K = 2*hi + v for row/col = lo.  So for point j (tile
// t=j>>4, slot ji=j&15) we store:
//   frag[t*32 + ji]      = K-pair (0,1)  for lanes 0-15
//   frag[t*32 + 16 + ji] = K-pair (2,3)  for lanes 16-31
// A-side contents: (z0,z1) / (z2,1).  B-side: (-2z0,-2z1) / (-2z2,||z||^2).
// ---------------------------------------------------------------------------
__global__ void __launch_bounds__(256)
chamfer_pack_kernel(const float* __restrict__ src,   // [count*3]
                    float2* __restrict__ fragA,
                    float2* __restrict__ fragB,
                    float*  __restrict__ norm,
                    int count)
{
    int j = blockIdx.x * blockDim.x + threadIdx.x;
    if (j >= count) return;
    float z0 = src[3 * j + 0];
    float z1 = src[3 * j + 1];
    float z2 = src[3 * j + 2];
    float n2 = z0 * z0 + z1 * z1 + z2 * z2;
    int base = (j >> 4) * 32 + (j & 15);
    fragA[base]      = make_float2(z0, z1);
    fragA[base + 16] = make_float2(z2, 1.0f);
    fragB[base]      = make_float2(-2.0f * z0, -2.0f * z1);
    fragB[base + 16] = make_float2(-2.0f * z2, n2);
    norm[j] = n2;
}

// ---------------------------------------------------------------------------
// Strip kernel: minsq[b, n] = min_m ||A-cloud[b,n] - B-cloud[b,m]||^2
//
// One workgroup (4 wave32) per 32 A-rows (two 16-row tiles sharing each B
// fragment). Each wave sweeps a quarter of the B tiles with a scalar trip
// count. Inner loop: 1 global b64 load, 2 v_wmma_f32_16x16x4_f32
// (D = A*B + C, C = ||x||^2 broadcast), 16 v_min_num_f32 (fast-math scoped so
// no sNaN canonicalize is inserted and VOPD dual-issue pairing is possible).
// ---------------------------------------------------------------------------
__global__ void __launch_bounds__(128)
chamfer_min_kernel(const v2f*   __restrict__ fragA, // [B*NP*2] lane-ordered
                   const float* __restrict__ normA, // [B*NP]
                   const v2f*   __restrict__ fragB, // [B*MP*2] lane-ordered
                   float* __restrict__ minsq,       // [B*NP] output
                   int NP, int MP)
{
    const int strips = NP >> 5;                 // 32 rows per workgroup
    const int b   = blockIdx.x / strips;
    const int sid = blockIdx.x % strips;
    const int n0  = sid << 5;

    const int tid  = threadIdx.x;
    const int wave = tid >> 5;
    const int lane = tid & 31;
    const int hi   = lane >> 4;

    const v2f*   fa = fragA + (size_t)b * NP * 2;
    const v2f*   fb = fragB + (size_t)b * MP * 2;
    const float* na = normA + (size_t)b * NP;

    // Two loop-invariant A fragments (rows n0..n0+15 and n0+16..n0+31).
    const v2f a0 = fa[(sid * 2 + 0) * 32 + lane];
    const v2f a1 = fa[(sid * 2 + 1) * 32 + lane];

    // Loop-invariant C initializers: ||x_i||^2, i = v + 8*hi (+16 for tile 1).
    v8f c0, c1;
#pragma unroll
    for (int v = 0; v < 8; ++v) {
        c0[v] = na[n0 + v + 8 * hi];
        c1[v] = na[n0 + 16 + v + 8 * hi];
    }

    float m0[8], m1[8];
#pragma unroll
    for (int v = 0; v < 8; ++v) { m0[v] = __builtin_inff(); m1[v] = __builtin_inff(); }

    // Scalar trip count: MP/16 tiles spread over 4 waves.
    // Keep the B-stream index in 32-bit so the loads use SADDR + voffset
    // (scale_offset) addressing with a cheap 32-bit increment.
    const int niter = MP >> 6;
    int idx = wave * 32 + lane;
    for (int it = 0; it < niter; ++it) {
        const v2f bf = fb[idx];
        idx += 128;                             // 4 tiles * 32 v2f

        v8f d0 = __builtin_amdgcn_wmma_f32_16x16x4_f32(
            false, a0, false, bf, (short)0, c0, false, false);
        v8f d1 = __builtin_amdgcn_wmma_f32_16x16x4_f32(
            false, a1, false, bf, (short)0, c1, false, false);

        {
#pragma float_control(precise, off)
            // nnan/nsz minnum: no canonicalize, VOPD-pairable v_min_num_f32.
#pragma unroll
            for (int v = 0; v < 8; ++v) {
                m0[v] = __builtin_fminf(m0[v], d0[v]);
                m1[v] = __builtin_fminf(m1[v], d1[v]);
            }
        }
    }

    // Cross-wave / cross-lane reduction of the 32 row minima via LDS.
    __shared__ float lds[4 * 2 * 8 * 32];
#pragma unroll
    for (int v = 0; v < 8; ++v) {
        lds[((wave * 2 + 0) * 8 + v) * 32 + lane] = m0[v];
        lds[((wave * 2 + 1) * 8 + v) * 32 + lane] = m1[v];
    }
    __syncthreads();

    if (tid < 32) {
        const int u  = tid >> 4;     // which 16-row tile
        const int ii = tid & 15;     // row within tile
        const int v  = ii & 7;
        const int h  = ii >> 3;
        float m = __builtin_inff();
        for (int w = 0; w < 4; ++w)
            for (int l = 0; l < 16; ++l)
                m = fminf(m, lds[((w * 2 + u) * 8 + v) * 32 + h * 16 + l]);
        minsq[(size_t)b * NP + n0 + tid] = m;
    }
}

// ---------------------------------------------------------------------------
// Finalize: loss = ( sum sqrt(min0) + sum sqrt(min1) ) / (B * 8192)
// (valid because N == M, so both means share the same normalizer)
// ---------------------------------------------------------------------------
__global__ void __launch_bounds__(256)
chamfer_finalize_kernel(const float* __restrict__ min0,
                        const float* __restrict__ min1,
                        float* __restrict__ out, int total)
{
    __shared__ float red[256];
    const int tid = threadIdx.x;
    float acc = 0.0f;
    for (int i = tid; i < total; i += 256)
        acc += sqrtf(min0[i]) + sqrtf(min1[i]);
    red[tid] = acc;
    __syncthreads();
    for (int s = 128; s > 0; s >>= 1) {
        if (tid < s) red[tid] += red[tid + s];
        __syncthreads();
    }
    if (tid == 0) out[0] = red[0] / (float)total;
}

// ---------------------------------------------------------------------------
extern "C" void kernel_launch(void* const* d_in, const int* in_sizes, int n_in,
                              void* d_out, int out_size, void* d_ws, size_t ws_size,
                              hipStream_t stream)
{
    (void)n_in; (void)out_size; (void)ws_size;
    const float* X = (const float*)d_in[0];   // [B, N, 3]
    const float* Y = (const float*)d_in[1];   // [B, M, 3]

    const int B = 4;
    const int N = in_sizes[0] / (B * 3);      // 8192
    const int M = in_sizes[1] / (B * 3);      // 8192

    const size_t PN = (size_t)B * N;
    const size_t PM = (size_t)B * M;

    // Workspace layout (floats): per cloud, fragA(4/pt) fragB(4/pt) norm(1/pt)
    float* ws = (float*)d_ws;
    float2* XfA  = (float2*)(ws);
    float2* XfB  = (float2*)(ws + 4 * PN);
    float*  Xn   =           ws + 8 * PN;
    float2* YfA  = (float2*)(ws + 9 * PN);
    float2* YfB  = (float2*)(ws + 9 * PN + 4 * PM);
    float*  Yn   =           ws + 9 * PN + 8 * PM;
    float*  min1 =           ws + 9 * PN + 9 * PM;        // [PN] min over m
    float*  min0 =           ws + 9 * PN + 9 * PM + PN;   // [PM] min over n

    chamfer_pack_kernel<<<(int)((PN + 255) / 256), 256, 0, stream>>>(X, XfA, XfB, Xn, (int)PN);
    chamfer_pack_kernel<<<(int)((PM + 255) / 256), 256, 0, stream>>>(Y, YfA, YfB, Yn, (int)PM);

    // dist1[b,n] = min_m d^2(x_n, y_m)
    chamfer_min_kernel<<<B * (N / 32), 128, 0, stream>>>(
        (const v2f*)XfA, Xn, (const v2f*)YfB, min1, N, M);
    // dist0[b,m] = min_n d^2(y_m, x_n)   (symmetric, roles swapped)
    chamfer_min_kernel<<<B * (M / 32), 128, 0, stream>>>(
        (const v2f*)YfA, Yn, (const v2f*)XfB, min0, M, N);

    chamfer_finalize_kernel<<<1, 256, 0, stream>>>(min0, min1, (float*)d_out,
                                                   (int)(PN + PM) / 2 /* B*8192 */);
}